// INGPTable_11991548690913
// MI455X (gfx1250) — compile-verified
//
#include <hip/hip_runtime.h>
#include <cstdint>
#include <cstddef>

// ---------------------------------------------------------------------------
// InstantNGP hash-grid lookup, MI455X (gfx1250, wave32).
//
//   out[i,f] = sum_{c in 8 corners} w_c * table[hash(corner_c), f]
//
// hash(c) = (c.x*1 ^ c.y*2654435761 ^ c.z*805459861) mod 2^19  -> pure u32,
// '& 0x7FFFF' (power-of-two modulus, only low 32 bits of products matter).
//
// Memory strategy (kernel is gather/L2 bound; VMEM issue is the scarce pipe):
//   * coords:  Tensor Data Mover DMA (tensor_load_to_lds) stages each block's
//              3072B coordinate tile into LDS on the independent TDM engine
//              (TENSORcnt), freeing the per-lane VMEM pipe for the gathers.
//              tensor_dim0 = remaining elems -> TDM zero-fills the tail OOB.
//              Wave 0 lanes also global_prefetch the tile 24 blocks ahead
//              (24 x 128B lines = one tile) to warm L2 for the DMA.
//   * table:   8x global_load_b64 gathers, default RT hint -> 4MB table stays
//              resident in the 192MB L2.
//   * output:  one nontemporal global_store_b64 per lane.
// ---------------------------------------------------------------------------

#define TABLE_MASK 0x7FFFFu      // 2^19 - 1
#define HASH_PI2   2654435761u
#define HASH_PI3   805459861u
#define RES_F      512.0f

#define BLOCK      256
#define TILE_F32   (BLOCK * 3)   // 768 floats = 3072 bytes per block tile
#define PF_BLOCKS  24            // prefetch distance in blocks

typedef float v3f __attribute__((ext_vector_type(3), aligned(4)));
typedef float v2f __attribute__((ext_vector_type(2), aligned(8)));
typedef unsigned int u32x4 __attribute__((ext_vector_type(4)));
typedef int i32x4 __attribute__((ext_vector_type(4)));
typedef int i32x8 __attribute__((ext_vector_type(8)));

#if defined(__has_builtin)
#if __has_builtin(__builtin_amdgcn_tensor_load_to_lds)
#define HAVE_TDM 1
#endif
#if __has_builtin(__builtin_amdgcn_s_wait_tensorcnt)
#define HAVE_TENSORCNT_BUILTIN 1
#endif
#endif

__device__ __forceinline__ void wait_tensorcnt_zero()
{
#if defined(HAVE_TENSORCNT_BUILTIN)
    __builtin_amdgcn_s_wait_tensorcnt(0);
#else
    asm volatile("s_wait_tensorcnt 0x0" ::: "memory");
#endif
}

#if defined(HAVE_TDM)
// 1-D (single-row tile) TDM load: tile_elems f32 from gsrc -> LDS byte addr.
// tensor_dim0 = tensor_elems bounds the read; OOB elements return zero.
__device__ __forceinline__ void tdm_load_1d_to_lds(uint32_t lds_byte_addr,
                                                   const float* gsrc,
                                                   uint32_t tensor_elems,
                                                   uint32_t tile_elems)
{
    const uint64_t ga = (uint64_t)(uintptr_t)gsrc;

    // D# group 0 (ISA 8.3): count=1 user descriptor, lds_addr, global_addr,
    // type=2 ("image").
    u32x4 g0;
    g0.x = 1u;
    g0.y = lds_byte_addr;
    g0.z = (uint32_t)ga;
    g0.w = ((uint32_t)(ga >> 32) & 0x01FFFFFFu) | (2u << 30);

    // D# group 1 (ISA 8.4): workgroup_mask=0 (not in cluster), data_size=2
    // (4 bytes), no barrier/iterate/pad; tensor_dim0=tensor_elems,
    // tensor_dim1=1, tile_dim0=tile_elems, tile_dim1/2=0 (unused -> 1-D).
    i32x8 g1;
    g1[0] = (int)(2u << 16);                            // data_size = 4B
    g1[1] = (int)((tensor_elems & 0xFFFFu) << 16);      // tensor_dim0[15:0]
    g1[2] = (int)((tensor_elems >> 16) | (1u << 16));   // dim0[31:16] | dim1=1
    g1[3] = (int)((tile_elems & 0xFFFFu) << 16);        // tile_dim0
    g1[4] = 0;                                          // tile_dim1/2 unused
    g1[5] = (int)tensor_elems;                          // dim0_stride[31:0]
    g1[6] = 0;                                          // stride hi, dim1_stride
    g1[7] = 0;

    const i32x4 z4 = {0, 0, 0, 0};
#if defined(__clang_major__) && (__clang_major__ >= 23)
    const i32x8 z8 = {0, 0, 0, 0, 0, 0, 0, 0};
    __builtin_amdgcn_tensor_load_to_lds(g0, g1, z4, z4, z8, 0);
#else
    __builtin_amdgcn_tensor_load_to_lds(g0, g1, z4, z4, 0);
#endif
}
#endif // HAVE_TDM

__global__ __launch_bounds__(BLOCK)
void ingp_hash_interp_kernel(const float* __restrict__ xp,
                             const float* __restrict__ table,
                             float* __restrict__ outp,
                             int n)
{
    const int tid  = (int)threadIdx.x;
    const int base = (int)blockIdx.x * BLOCK;
    const int i    = base + tid;

#if defined(HAVE_TDM)
    __shared__ float smem[TILE_F32];

    if (tid < 32) {  // wave 0 drives the block's TDM DMA (EXEC-uniform branch)
        // Warm L2 for the tile PF_BLOCKS ahead: one 128B line per lane.
        const int pf_base = base + PF_BLOCKS * BLOCK;
        if (pf_base < n && tid < (TILE_F32 * 4) / 128) {
            __builtin_prefetch((const char*)xp + (size_t)pf_base * 12 +
                                   (size_t)tid * 128, 0, 1);
        }
        const uint32_t lds_addr = (uint32_t)(uintptr_t)(void*)&smem[0];
        const uint32_t rem_elems = (uint32_t)(n - base) * 3u;
        tdm_load_1d_to_lds(lds_addr, xp + (size_t)3 * (size_t)base,
                           rem_elems, TILE_F32);
        wait_tensorcnt_zero();   // per-wave TENSORcnt -> 0 BEFORE the barrier
    }
    __syncthreads();             // publish LDS tile to all 8 waves
#endif

    if (i >= n) return;          // (after the barrier: no wave stalls it)

#if defined(HAVE_TDM)
    // Conflict-free LDS read-back: dword addresses stride-3 across 64 banks.
    const v3f p = *(const v3f*)(smem + 3 * tid);
#else
    const v3f p = __builtin_nontemporal_load(
        (const v3f*)(xp + (size_t)3 * (size_t)i));
#endif

    const float sx = p.x * RES_F;
    const float sy = p.y * RES_F;
    const float sz = p.z * RES_F;

    const float fx = floorf(sx);
    const float fy = floorf(sy);
    const float fz = floorf(sz);

    // Corner integer coords (0..512).
    const uint32_t ix = (uint32_t)(int)fx;
    const uint32_t iy = (uint32_t)(int)fy;
    const uint32_t iz = (uint32_t)(int)fz;

    // Per-dimension weights, arithmetic identical to the reference:
    //   w(off) = 1 - |xs - (floor + off)|
    const float wx0 = 1.0f - __builtin_fabsf(sx - fx);
    const float wx1 = 1.0f - __builtin_fabsf(sx - (fx + 1.0f));
    const float wy0 = 1.0f - __builtin_fabsf(sy - fy);
    const float wy1 = 1.0f - __builtin_fabsf(sy - (fy + 1.0f));
    const float wz0 = 1.0f - __builtin_fabsf(sz - fz);
    const float wz1 = 1.0f - __builtin_fabsf(sz - (fz + 1.0f));

    // Hash partial products shared across corners (4 v_mul_lo_u32 total).
    const uint32_t hy0 = iy * HASH_PI2;
    const uint32_t hy1 = (iy + 1u) * HASH_PI2;
    const uint32_t hz0 = iz * HASH_PI3;
    const uint32_t hz1 = (iz + 1u) * HASH_PI3;
    const uint32_t ix1 = ix + 1u;

    // All 8 indices before any gather, so the 8 b64 loads clause together.
    const uint32_t i000 = (ix  ^ hy0 ^ hz0) & TABLE_MASK;
    const uint32_t i001 = (ix  ^ hy0 ^ hz1) & TABLE_MASK;
    const uint32_t i010 = (ix  ^ hy1 ^ hz0) & TABLE_MASK;
    const uint32_t i011 = (ix  ^ hy1 ^ hz1) & TABLE_MASK;
    const uint32_t i100 = (ix1 ^ hy0 ^ hz0) & TABLE_MASK;
    const uint32_t i101 = (ix1 ^ hy0 ^ hz1) & TABLE_MASK;
    const uint32_t i110 = (ix1 ^ hy1 ^ hz0) & TABLE_MASK;
    const uint32_t i111 = (ix1 ^ hy1 ^ hz1) & TABLE_MASK;

    // ---- 8 gathers of float2 feature pairs (global_load_b64, RT hint) ------
    const v2f* __restrict__ tp = (const v2f*)table;
    const v2f f000 = tp[i000];
    const v2f f001 = tp[i001];
    const v2f f010 = tp[i010];
    const v2f f011 = tp[i011];
    const v2f f100 = tp[i100];
    const v2f f101 = tp[i101];
    const v2f f110 = tp[i110];
    const v2f f111 = tp[i111];

    // Trilinear weights, multiplied in the reference's (x*y)*z order.
    const float wx0y0 = wx0 * wy0;
    const float wx0y1 = wx0 * wy1;
    const float wx1y0 = wx1 * wy0;
    const float wx1y1 = wx1 * wy1;

    // Accumulate in reference corner order (z fastest).
    v2f acc = f000 * (wx0y0 * wz0);
    acc += f001 * (wx0y0 * wz1);
    acc += f010 * (wx0y1 * wz0);
    acc += f011 * (wx0y1 * wz1);
    acc += f100 * (wx1y0 * wz0);
    acc += f101 * (wx1y0 * wz1);
    acc += f110 * (wx1y1 * wz0);
    acc += f111 * (wx1y1 * wz1);

    // ---- streamed output, 8B per lane (nontemporal b64 store) --------------
    __builtin_nontemporal_store(acc, (v2f*)outp + i);
}

extern "C" void kernel_launch(void* const* d_in, const int* in_sizes, int n_in,
                              void* d_out, int out_size, void* d_ws, size_t ws_size,
                              hipStream_t stream)
{
    const float* x     = (const float*)d_in[0];   // [N,3] f32
    const float* table = (const float*)d_in[1];   // [2^19, 2] f32
    float*       out   = (float*)d_out;           // [N,2] f32

    const int n = in_sizes[0] / 3;

    int grid = (n + BLOCK - 1) / BLOCK;           // one point per thread
    if (grid < 1) grid = 1;

    ingp_hash_interp_kernel<<<grid, BLOCK, 0, stream>>>(x, table, out, n);
}